// MultiBoxLoss_65326452572232
// MI455X (gfx1250) — compile-verified
//
#include <hip/hip_runtime.h>
#include <math.h>

#define THRESH     0.5f
#define NEG_RATIO  3
#define CVAR       0.1f
#define SVAR       0.2f
#define GMAX       16
#define CE_TILE    128      // priors staged per block in k_ce
#define MINE_CAP   40       // supports P <= 40*256 (P = 8732)

// ---- CDNA5 async global->LDS probe ------------------------------------
#if defined(__has_builtin)
#  if __has_builtin(__builtin_amdgcn_global_load_async_to_lds_b128)
#    define HAVE_ASYNC 1
#  endif
#endif
#ifndef HAVE_ASYNC
#  define HAVE_ASYNC 0
#endif

typedef int v4i_t __attribute__((ext_vector_type(4)));
typedef __attribute__((address_space(1))) v4i_t gv4i;   // global int4
typedef __attribute__((address_space(3))) v4i_t lv4i;   // LDS int4

__device__ __forceinline__ void wait_async0() {
#if defined(__has_builtin) && __has_builtin(__builtin_amdgcn_s_wait_asynccnt)
  __builtin_amdgcn_s_wait_asynccnt(0);
#else
  asm volatile("s_wait_asynccnt 0" ::: "memory");
#endif
}

// ---- order-preserving float<->uint mapping (for radix top-K select) ---
__device__ __forceinline__ unsigned mapf(float v) {
  unsigned u = __float_as_uint(v);
  return (u & 0x80000000u) ? ~u : (u | 0x80000000u);
}
__device__ __forceinline__ float unmapf(unsigned m) {
  unsigned u = (m & 0x80000000u) ? (m ^ 0x80000000u) : ~m;
  return __uint_as_float(u);
}

// ======================================================================
__global__ void k_init(int* __restrict__ npos, int* __restrict__ total, int B) {
  int t = blockIdx.x * blockDim.x + threadIdx.x;
  if (t < B) npos[t] = 0;
  if (t == 0) total[0] = 0;
}

// One block per image: per-prior best GT, and per-GT best prior (argmax over P).
__global__ void k_match1(const float* __restrict__ gt_boxes,   // [B,G,4] corner
                         const float* __restrict__ priors,     // [P,4] center
                         float* __restrict__ best_ov,          // [B,P]
                         int*   __restrict__ best_idx,         // [B,P]
                         int*   __restrict__ best_prior,       // [B,G]
                         int P, int G) {
  const int b = blockIdx.x;
  const int t = threadIdx.x;              // blockDim.x == 256
  __shared__ float s_gt[GMAX * 4];
  __shared__ float s_ga[GMAX];
  if (t < G * 4) s_gt[t] = gt_boxes[(size_t)b * G * 4 + t];
  __syncthreads();
  if (t < G) {
    float x0 = s_gt[t*4], y0 = s_gt[t*4+1], x1 = s_gt[t*4+2], y1 = s_gt[t*4+3];
    s_ga[t] = (x1 - x0) * (y1 - y0);
  }
  __syncthreads();

  float gb_ov[GMAX];
  int   gb_p[GMAX];
#pragma unroll
  for (int g = 0; g < GMAX; ++g) { gb_ov[g] = -1.0f; gb_p[g] = 0x7fffffff; }

  for (int p = t; p < P; p += 256) {
    float cx = priors[p*4+0], cy = priors[p*4+1];
    float w  = priors[p*4+2], h  = priors[p*4+3];
    float px0 = cx - 0.5f*w, py0 = cy - 0.5f*h;
    float px1 = cx + 0.5f*w, py1 = cy + 0.5f*h;
    float ap  = w * h;
    float bov = -1.0f; int bg = 0;
#pragma unroll
    for (int g = 0; g < GMAX; ++g) {
      if (g < G) {
        float lx = fmaxf(s_gt[g*4+0], px0), ly = fmaxf(s_gt[g*4+1], py0);
        float rx = fminf(s_gt[g*4+2], px1), ry = fminf(s_gt[g*4+3], py1);
        float iw = fmaxf(rx - lx, 0.0f), ih = fmaxf(ry - ly, 0.0f);
        float inter = iw * ih;
        float ov = inter / (s_ga[g] + ap - inter);
        if (ov > bov) { bov = ov; bg = g; }           // first max -> lowest g
        if (ov > gb_ov[g]) { gb_ov[g] = ov; gb_p[g] = p; } // lowest p kept
      }
    }
    best_ov[(size_t)b*P + p]  = bov;
    best_idx[(size_t)b*P + p] = bg;
  }
  __syncthreads();

  __shared__ float r_ov[256];
  __shared__ int   r_p[256];
  for (int g = 0; g < G; ++g) {
    r_ov[t] = gb_ov[g]; r_p[t] = gb_p[g];
    __syncthreads();
    for (int s = 128; s > 0; s >>= 1) {
      if (t < s) {
        float o2 = r_ov[t+s]; int p2 = r_p[t+s];
        if (o2 > r_ov[t] || (o2 == r_ov[t] && p2 < r_p[t])) { r_ov[t] = o2; r_p[t] = p2; }
      }
      __syncthreads();
    }
    if (t == 0) best_prior[(size_t)b*G + g] = r_p[0];
    __syncthreads();
  }
}

// Force-match, threshold -> conf; encode + masked L1 partials; n_pos counts.
__global__ void k_match2(const float* __restrict__ pred_boxes, // [B,P,4]
                         const float* __restrict__ gt_boxes,   // [B,G,4]
                         const int*   __restrict__ gt_labels,  // [B,G]
                         const float* __restrict__ priors,     // [P,4]
                         const float* __restrict__ best_ov,
                         const int*   __restrict__ best_idx,
                         const int*   __restrict__ best_prior, // [B,G]
                         int*   __restrict__ conf_out,         // [B,P]
                         float* __restrict__ l1_part,          // [gridDim.x]
                         int*   __restrict__ npos,             // [B]
                         int*   __restrict__ total_pos,        // [1]
                         int P, int G, int blocksPerImage) {
  const int b = blockIdx.x / blocksPerImage;
  const int p = (blockIdx.x % blocksPerImage) * blockDim.x + threadIdx.x;
  float l1 = 0.0f; int posc = 0;
  if (p < P) {
    float ov = best_ov[(size_t)b*P + p];
    int   g  = best_idx[(size_t)b*P + p];
    for (int k = 0; k < G; ++k)
      if (best_prior[(size_t)b*G + k] == p) { ov = 2.0f; g = k; }  // last g wins
    int conf = gt_labels[(size_t)b*G + g];
    if (ov < THRESH) conf = 0;
    conf_out[(size_t)b*P + p] = conf;
    if (conf != 0) {
      posc = 1;
      const float* m = &gt_boxes[((size_t)b*G + g) * 4];
      float mcx = (m[0] + m[2]) * 0.5f, mcy = (m[1] + m[3]) * 0.5f;
      float mw  = m[2] - m[0],          mh  = m[3] - m[1];
      float pcx = priors[p*4+0], pcy = priors[p*4+1];
      float pw  = priors[p*4+2], ph  = priors[p*4+3];
      float e0 = (mcx - pcx) / (CVAR * pw);
      float e1 = (mcy - pcy) / (CVAR * ph);
      float e2 = logf(mw / pw) / SVAR;
      float e3 = logf(mh / ph) / SVAR;
      const float* pb = &pred_boxes[((size_t)b*P + p) * 4];
      l1 = fabsf(pb[0]-e0) + fabsf(pb[1]-e1) + fabsf(pb[2]-e2) + fabsf(pb[3]-e3);
    }
  }
  __shared__ float sl[256];
  __shared__ int   sc[256];
  sl[threadIdx.x] = l1; sc[threadIdx.x] = posc;
  __syncthreads();
  for (int s = 128; s > 0; s >>= 1) {
    if (threadIdx.x < s) { sl[threadIdx.x] += sl[threadIdx.x+s]; sc[threadIdx.x] += sc[threadIdx.x+s]; }
    __syncthreads();
  }
  if (threadIdx.x == 0) {
    l1_part[blockIdx.x] = sl[0];
    if (sc[0]) { atomicAdd(&npos[b], sc[0]); atomicAdd(total_pos, sc[0]); }
  }
}

// Bandwidth-dominant kernel: stream scores via async global->LDS tiles,
// per-prior log-softmax CE; positives accumulate pos_loss, negatives -> ce_out.
__global__ void k_ce(const float* __restrict__ scores,   // [B,P,C]
                     const int*   __restrict__ conf,     // [B,P]
                     float* __restrict__ ce_out,         // [B,P] (negatives, 0 at pos)
                     float* __restrict__ pos_part,       // [gridDim.x]
                     int P, int C, int tilesPerImage) {
  __shared__ __align__(16) float s_sc[CE_TILE * 81 + 8];
  const int b    = blockIdx.x / tilesPerImage;
  const int base = (blockIdx.x % tilesPerImage) * CE_TILE;
  int n = P - base; if (n > CE_TILE) n = CE_TILE;
  const int t   = threadIdx.x;                 // blockDim.x == CE_TILE
  const int nfl = n * C;
  const size_t gbyte = ((size_t)b * P + base) * (size_t)C * 4;

#if HAVE_ASYNC
  const size_t al   = gbyte & ~(size_t)15;
  const int shift   = (int)(gbyte - al);       // 0/4/8/12
  const int n16     = (nfl * 4 + shift + 15) >> 4;
  gv4i* gp = (gv4i*)((size_t)scores + al);     // AS1 = full 64-bit address
  lv4i* lp = (lv4i*)&s_sc[0];                  // generic -> AS3 addrspacecast
  for (int i = t; i < n16; i += CE_TILE)
    __builtin_amdgcn_global_load_async_to_lds_b128(gp + i, lp + i, 0, 0);
  wait_async0();
  __syncthreads();
  const float* sp0 = s_sc + (shift >> 2);
#else
  const float* gsrc = (const float*)((const char*)scores + gbyte);
  for (int i = t; i < nfl; i += CE_TILE) s_sc[i] = gsrc[i];
  __syncthreads();
  const float* sp0 = s_sc;
#endif

  float ploc = 0.0f;
  if (t < n) {
    const float* sp = sp0 + t * C;   // stride 81 -> conflict-free (gcd(81,64)=1)
    float m = sp[0];
    for (int c = 1; c < C; ++c) m = fmaxf(m, sp[c]);
    float se = 0.0f;
    for (int c = 0; c < C; ++c) se += expf(sp[c] - m);
    float lse = logf(se) + m;
    const int cf = conf[(size_t)b*P + base + t];
    float ce = lse - sp[cf];
    if (cf != 0) { ploc = ce; ce = 0.0f; }
    ce_out[(size_t)b*P + base + t] = ce;
  }
  __shared__ float sred[CE_TILE];
  sred[t] = ploc;
  __syncthreads();
  for (int s = CE_TILE/2; s > 0; s >>= 1) { if (t < s) sred[t] += sred[t+s]; __syncthreads(); }
  if (t == 0) pos_part[blockIdx.x] = sred[0];
}

// Per image: exact top-K sum via 32-step bitwise radix-select (K = 3*n_pos).
__global__ void k_mine(const float* __restrict__ ce,    // [B,P]
                       const int*   __restrict__ npos,  // [B]
                       float* __restrict__ neg_part,    // [B]
                       int P) {
  const int b = blockIdx.x, t = threadIdx.x;            // blockDim.x == 256
  int K = NEG_RATIO * npos[b];
  if (K > P) K = P;

  float    vv[MINE_CAP];
  unsigned mb[MINE_CAP];
#pragma unroll
  for (int i = 0; i < MINE_CAP; ++i) {
    int p = t + i * 256;
    float v = (p < P) ? ce[(size_t)b*P + p] : 0.0f;    // pad 0: harmless (v >= 0)
    vv[i] = v; mb[i] = mapf(v);
  }

  __shared__ int   sredi[256];
  __shared__ float sredf[256];

  unsigned thr = 0u;
  if (K > 0 && K < P) {
    for (int bit = 31; bit >= 0; --bit) {
      const unsigned cand = thr | (1u << bit);
      int c = 0;
#pragma unroll
      for (int i = 0; i < MINE_CAP; ++i) c += (mb[i] >= cand) ? 1 : 0;
      sredi[t] = c;
      __syncthreads();
      for (int s = 128; s > 0; s >>= 1) { if (t < s) sredi[t] += sredi[t+s]; __syncthreads(); }
      const int total = sredi[0];
      __syncthreads();
      if (total >= K) thr = cand;
    }
  }

  float s = 0.0f; int cgt = 0;
  if (K >= P) {
#pragma unroll
    for (int i = 0; i < MINE_CAP; ++i) s += vv[i];
  } else if (K > 0) {
#pragma unroll
    for (int i = 0; i < MINE_CAP; ++i) if (mb[i] > thr) { s += vv[i]; ++cgt; }
  }
  sredf[t] = s; sredi[t] = cgt;
  __syncthreads();
  for (int st = 128; st > 0; st >>= 1) {
    if (t < st) { sredf[t] += sredf[t+st]; sredi[t] += sredi[t+st]; }
    __syncthreads();
  }
  if (t == 0) {
    float S = sredf[0];
    if (K > 0 && K < P) S += (float)(K - sredi[0]) * unmapf(thr);
    neg_part[b] = S;
  }
}

__global__ void k_final(const float* __restrict__ l1_part, int n1,
                        const float* __restrict__ pos_part, int n2,
                        const float* __restrict__ neg_part, int nB,
                        const int*   __restrict__ total_pos,
                        float* __restrict__ out) {
  __shared__ float s1[256], s2[256], s3[256];
  const int t = threadIdx.x;
  float a = 0.0f, b = 0.0f, c = 0.0f;
  for (int i = t; i < n1; i += 256) a += l1_part[i];
  for (int i = t; i < n2; i += 256) b += pos_part[i];
  for (int i = t; i < nB; i += 256) c += neg_part[i];
  s1[t] = a; s2[t] = b; s3[t] = c;
  __syncthreads();
  for (int s = 128; s > 0; s >>= 1) {
    if (t < s) { s1[t] += s1[t+s]; s2[t] += s2[t+s]; s3[t] += s3[t+s]; }
    __syncthreads();
  }
  if (t == 0) {
    const float tp = (float)total_pos[0];
    out[0] = (s2[0] + s3[0]) / tp;        // confidence_loss
    out[1] = s1[0] / (tp * 4.0f);         // ALPHA * location_loss, ALPHA=1
  }
}

// ======================================================================
extern "C" void kernel_launch(void* const* d_in, const int* in_sizes, int n_in,
                              void* d_out, int out_size, void* d_ws, size_t ws_size,
                              hipStream_t stream) {
  const float* pred_boxes  = (const float*)d_in[0];
  const float* pred_scores = (const float*)d_in[1];
  const float* gt_boxes    = (const float*)d_in[2];
  const int*   gt_labels   = (const int*)d_in[3];
  const float* priors      = (const float*)d_in[4];

  const int P = in_sizes[4] / 4;
  const int B = in_sizes[0] / (P * 4);
  const int C = in_sizes[1] / (B * P);
  const int G = in_sizes[3] / B;

  const int m2Blocks      = (P + 255) / 256;
  const int tilesPerImage = (P + CE_TILE - 1) / CE_TILE;
  const int ceBlocks      = B * tilesPerImage;

  size_t off = 0;
  auto carve = [&](size_t bytes) -> void* {
    void* p = (char*)d_ws + off;
    off += (bytes + 255) & ~(size_t)255;
    return p;
  };
  float* ws_best_ov    = (float*)carve((size_t)B * P * 4);
  int*   ws_best_idx   = (int*)  carve((size_t)B * P * 4);
  int*   ws_conf       = (int*)  carve((size_t)B * P * 4);
  float* ws_ce         = (float*)carve((size_t)B * P * 4);
  int*   ws_best_prior = (int*)  carve((size_t)B * G * 4);
  int*   ws_npos       = (int*)  carve((size_t)B * 4);
  int*   ws_total      = (int*)  carve(4);
  float* ws_l1part     = (float*)carve((size_t)B * m2Blocks * 4);
  float* ws_pospart    = (float*)carve((size_t)ceBlocks * 4);
  float* ws_negpart    = (float*)carve((size_t)B * 4);
  (void)ws_size; (void)n_in; (void)out_size;

  k_init<<<1, 256, 0, stream>>>(ws_npos, ws_total, B);
  k_match1<<<B, 256, 0, stream>>>(gt_boxes, priors, ws_best_ov, ws_best_idx,
                                  ws_best_prior, P, G);
  k_match2<<<B * m2Blocks, 256, 0, stream>>>(pred_boxes, gt_boxes, gt_labels, priors,
                                             ws_best_ov, ws_best_idx, ws_best_prior,
                                             ws_conf, ws_l1part, ws_npos, ws_total,
                                             P, G, m2Blocks);
  k_ce<<<ceBlocks, CE_TILE, 0, stream>>>(pred_scores, ws_conf, ws_ce, ws_pospart,
                                         P, C, tilesPerImage);
  k_mine<<<B, 256, 0, stream>>>(ws_ce, ws_npos, ws_negpart, P);
  k_final<<<1, 256, 0, stream>>>(ws_l1part, B * m2Blocks, ws_pospart, ceBlocks,
                                 ws_negpart, B, ws_total, (float*)d_out);
}